// Encoder_Postnet_15367392985793
// MI455X (gfx1250) — compile-verified
//
#include <hip/hip_runtime.h>

// Problem constants (fixed by the reference).
constexpr int B       = 32;
constexpr int T_TEXT  = 512;
constexpr int T_ALIGN = 4096;
constexpr int D       = 512;
constexpr int D4      = D / 4;          // float4 chunks per row (128)
constexpr int FPB     = 8;              // frames per block (one wave per frame)

#define AS1 __attribute__((address_space(1)))
#define AS3 __attribute__((address_space(3)))

// Matches the builtin's parameter type reported by clang:
//   __attribute__((__vector_size__(4 * sizeof(int)))) int *
typedef int v4i __attribute__((__vector_size__(16)));

// gfx12-family CPol: TH in bits [2:0]; TH=1 -> non-temporal.
#define CPOL_RT 0
#define CPOL_NT 1

#if defined(__AMDGCN__) &&                                                  \
    __has_builtin(__builtin_amdgcn_global_load_async_to_lds_b128) &&        \
    __has_builtin(__builtin_amdgcn_global_store_async_from_lds_b128)
#define USE_ASYNC_LDS 1
#else
#define USE_ASYNC_LDS 0
#endif

__device__ __forceinline__ void wait_async0() {
#if defined(__AMDGCN__)
#if __has_builtin(__builtin_amdgcn_s_wait_asynccnt)
  __builtin_amdgcn_s_wait_asynccnt(0);
#else
  asm volatile("s_wait_asynccnt 0" ::: "memory");
#endif
#endif
}

// ---------------------------------------------------------------------------
// Kernel 1: exact sequential alignment walk. One lane per batch element; all
// 32 batches run in parallel across the lanes of a single wave32. text rows
// are staged in LDS; `nxt` pre-loads text[ind+1] so the loop-carried chain is
// just a handful of VALU selects (the LDS load latency only matters on
// back-to-back mismatches, which real duration-expanded data doesn't have).
// Writes idx (or -1 when the walk has overflowed) into the workspace.
// ---------------------------------------------------------------------------
__global__ __launch_bounds__(32) void align_scan_kernel(
    const int* __restrict__ align, const int* __restrict__ text,
    int* __restrict__ idxws) {
  __shared__ int tex[B * T_TEXT];  // 64 KB of the 320 KB WGP LDS
  for (int i = threadIdx.x; i < B * T_TEXT; i += 32) tex[i] = text[i];
  __syncthreads();

  const int b = threadIdx.x;
  const int* __restrict__ arow = align + b * T_ALIGN;
  const int* __restrict__ trow = tex + b * T_TEXT;
  int* __restrict__ orow = idxws + b * T_ALIGN;

  int ind    = 0;
  int before = trow[0];
  int nxt    = trow[1];     // invariant: nxt == text[min(ind+1, T_TEXT-1)]
  int done   = 0;
  orow[0] = 0;              // frame 0 always copies encoder row 0

  for (int t = 1; t < T_ALIGN; ++t) {
    const int a     = arow[t];
    const int match = (a == before);
    const int inc   = ind + (match ? 0 : 1);
    const int ovf   = (inc >= T_TEXT);
    const int nd    = done | ((!match) & ovf);
    const int adv   = (!match) & (!nd);
    ind    = adv ? inc : ind;
    before = adv ? nxt : before;
    const int nsafe = (ind + 1 < T_TEXT) ? (ind + 1) : (T_TEXT - 1);
    nxt  = adv ? trow[nsafe] : nxt;   // latency hidden until next mismatch
    done = nd;
    orow[t] = nd ? -1 : ind;
  }
}

// ---------------------------------------------------------------------------
// Kernel 2: bandwidth-bound gather/expand of 256 MB.
// grid = (T_ALIGN/8, B), 256 threads = 8 waves; wave w of a block owns frame
// t = blockIdx.x*8 + w, so the row index is wave-uniform (no divergence).
// Each lane moves 4 x 16B through LDS with the CDNA5 async copy engine:
//   global_load_async_to_lds_b128 (RT: keep enc rows hot in L2 for 8x reuse)
//   s_wait_asynccnt 0
//   global_store_async_from_lds_b128 (NT: 256 MB write-once stream, don't
//                                     thrash the 192 MB L2 / evict enc)
// Each lane stores exactly the LDS bytes it loaded, so the wave-level
// ASYNCcnt wait is the only synchronization needed (no barriers).
// ---------------------------------------------------------------------------
__global__ __launch_bounds__(256) void gather_expand_kernel(
    const float* __restrict__ enc, const int* __restrict__ idxws,
    float* __restrict__ out) {
  const int wave = threadIdx.x >> 5;
  const int lane = threadIdx.x & 31;
  const int t    = blockIdx.x * FPB + wave;
  const int b    = blockIdx.y;

  const int idx = idxws[b * T_ALIGN + t];  // wave-uniform
  const float4* enc4 = (const float4*)enc;
  float4* out4 = (float4*)out;
  const long dstBase = (long)(b * T_ALIGN + t) * D4;

#if USE_ASYNC_LDS
  __shared__ float4 stage[FPB * D4];  // 16 KB staging buffer
  if (idx >= 0) {
    const long srcBase = (long)(b * T_TEXT + idx) * D4;
#pragma unroll
    for (int j = 0; j < 4; ++j) {
      const int c = lane + 32 * j;
      __builtin_amdgcn_global_load_async_to_lds_b128(
          (AS1 v4i*)(enc4 + srcBase + c),
          (AS3 v4i*)(&stage[wave * D4 + c]), 0, CPOL_RT);
    }
    wait_async0();  // async LDS writes -> async LDS reads are unordered
#pragma unroll
    for (int j = 0; j < 4; ++j) {
      const int c = lane + 32 * j;
      __builtin_amdgcn_global_store_async_from_lds_b128(
          (AS1 v4i*)(out4 + dstBase + c),
          (AS3 v4i*)(&stage[wave * D4 + c]), 0, CPOL_NT);
    }
    wait_async0();  // drain before LDS/wave teardown
  } else {
    const float4 z = make_float4(0.f, 0.f, 0.f, 0.f);
#pragma unroll
    for (int j = 0; j < 4; ++j) out4[dstBase + lane + 32 * j] = z;
  }
#else
  // Fallback: direct B128 copy through VGPRs (plus gfx1250 prefetch).
  if (idx >= 0) {
    const long srcBase = (long)(b * T_TEXT + idx) * D4;
    __builtin_prefetch(enc4 + srcBase + lane, 0, 0);
#pragma unroll
    for (int j = 0; j < 4; ++j) {
      const int c = lane + 32 * j;
      out4[dstBase + c] = enc4[srcBase + c];
    }
  } else {
    const float4 z = make_float4(0.f, 0.f, 0.f, 0.f);
#pragma unroll
    for (int j = 0; j < 4; ++j) out4[dstBase + lane + 32 * j] = z;
  }
#endif
}

// ---------------------------------------------------------------------------
// Inputs (setup_inputs order): encoder_out f32[32,512,512],
//                              align_phone i32[32,4096],
//                              text_phone  i32[32,512]
// Output: f32[32,4096,512]. Workspace: 512 KB of int idx values.
// ---------------------------------------------------------------------------
extern "C" void kernel_launch(void* const* d_in, const int* in_sizes, int n_in,
                              void* d_out, int out_size, void* d_ws,
                              size_t ws_size, hipStream_t stream) {
  (void)in_sizes; (void)n_in; (void)out_size; (void)ws_size;
  const float* enc  = (const float*)d_in[0];
  const int* align  = (const int*)d_in[1];
  const int* text   = (const int*)d_in[2];
  float* out        = (float*)d_out;
  int* idxws        = (int*)d_ws;  // B*T_ALIGN ints = 512 KB

  align_scan_kernel<<<1, 32, 0, stream>>>(align, text, idxws);

  dim3 grid(T_ALIGN / FPB, B);
  gather_expand_kernel<<<grid, 256, 0, stream>>>(enc, idxws, out);
}